// TsSub_29257317220823
// MI455X (gfx1250) — compile-verified
//
#include <hip/hip_runtime.h>

typedef float v2f __attribute__((ext_vector_type(2)));
typedef float v8f __attribute__((ext_vector_type(8)));

#define B_     256
#define F_     64
#define T_     2000
#define OUT_T  200      // (2000-1)/10 + 1
#define NPAIR  2016     // 64*63/2
#define TTILE  16
#define NTTILE 13       // ceil(200/16)
#define NPTILE 126      // 2016/16

__global__ __launch_bounds__(256)
void ts_pairdiff_wmma(const float* __restrict__ x, float* __restrict__ out) {
  // Staged subsampled tile: Xs[f][tt], tt fastest. 64*16*4B = 4KB of the 320KB/WGP LDS.
  __shared__ float Xs[F_ * TTILE];

  const int b   = blockIdx.y;
  const int t0  = blockIdx.x * TTILE;
  const int tid = threadIdx.x;

  // Stage: Xs[f][tt] = x[b, f, (t0+tt)*10]; zero-fill past t=199 (masked on store).
  for (int idx = tid; idx < F_ * TTILE; idx += 256) {
    int f  = idx >> 4;
    int tt = idx & 15;
    int t  = t0 + tt;
    float v = 0.0f;
    if (t < OUT_T)
      v = x[(size_t)b * (F_ * T_) + (size_t)f * T_ + (size_t)t * 10];
    Xs[idx] = v;
  }
  __syncthreads();

  const int wave = tid >> 5;   // 8 waves per block (wave32)
  const int lane = tid & 31;
  const int half = lane >> 4;  // 0: K={0,1} / rows M=0..7 of C ; 1: K={2,3} / M=8..15
  const int l16  = lane & 15;

  // Each wave handles p-tiles wave, wave+8, ... (uniform per wave -> EXEC all 1s at WMMA)
  for (int pt = wave; pt < NPTILE; pt += 8) {
    // This lane's A-matrix row = pair index p; decode triu (i,j) from p.
    int p = pt * TTILE + l16;
    int i = 0, rem = p;
    while (rem >= 63 - i) { rem -= 63 - i; ++i; }
    int j = i + 1 + rem;

    v8f acc = {};
#pragma unroll
    for (int kc = 0; kc < 16; ++kc) {
      // A 16x4 f32 layout: lanes 0-15 hold K=4kc+{0,1}, lanes 16-31 hold K=4kc+{2,3}, row M=l16
      // B 4x16 f32 layout: lanes 0-15 hold rows K=4kc+{0,1} at N=l16, lanes 16-31 rows K=4kc+{2,3}
      int f0 = kc * 4 + half * 2;
      v2f a, bb;
      a.x = (f0     == i) ? 1.0f : ((f0     == j) ? -1.0f : 0.0f);
      a.y = (f0 + 1 == i) ? 1.0f : ((f0 + 1 == j) ? -1.0f : 0.0f);
      bb.x = Xs[f0 * TTILE + l16];
      bb.y = Xs[(f0 + 1) * TTILE + l16];
      // D = A*B + C, exact in f32 since A entries are {+1,-1,0}
      acc = __builtin_amdgcn_wmma_f32_16x16x4_f32(false, a, false, bb,
                                                  (short)0, acc, false, false);
    }

    // C/D layout: VGPR r -> M = r + 8*half, N = l16. out[b, pt*16+M, t0+N]
    int t = t0 + l16;
    if (t < OUT_T) {
      size_t base = (size_t)b * (NPAIR * OUT_T) + (size_t)t;
#pragma unroll
      for (int r = 0; r < 8; ++r) {
        int pp = pt * TTILE + half * 8 + r;
        // streaming write-once output: non-temporal to spare L2
        __builtin_nontemporal_store(acc[r], &out[base + (size_t)pp * OUT_T]);
      }
    }
  }
}

extern "C" void kernel_launch(void* const* d_in, const int* in_sizes, int n_in,
                              void* d_out, int out_size, void* d_ws, size_t ws_size,
                              hipStream_t stream) {
  const float* x = (const float*)d_in[0];
  float* out = (float*)d_out;
  dim3 grid(NTTILE, B_);   // 13 t-tiles x 256 batches = 3328 blocks, 8 waves each
  ts_pairdiff_wmma<<<grid, 256, 0, stream>>>(x, out);
}